// OctreeInterp_77472620085713
// MI455X (gfx1250) — compile-verified
//
#include <hip/hip_runtime.h>
#include <stdint.h>

#define TPB 256
#define COARSE_MAX 4096

typedef float v4f __attribute__((ext_vector_type(4)));

// Low 32 bits of a flat shared-memory pointer are the LDS byte offset (CDNA5
// flat-aperture mapping).
__device__ __forceinline__ uint32_t lds_offset(const void* p) {
    return (uint32_t)(uintptr_t)p;
}

// ---------------- LUT build (only runs if d_ws can hold R^3 ints) -----------
__global__ void octree_fill_lut(int* __restrict__ inv, const int* __restrict__ depth_p,
                                unsigned long long ws_bytes) {
    int depth = *depth_p;
    if (3 * depth > 30) return;
    long long n = 1ll << (3 * depth);
    if ((unsigned long long)n * 4ull > ws_bytes) return;
    long long stride = (long long)gridDim.x * blockDim.x;
    for (long long i = (long long)blockIdx.x * blockDim.x + threadIdx.x; i < n; i += stride)
        inv[i] = -1;
}

__global__ void octree_scatter_lut(int* __restrict__ inv, const int* __restrict__ keys,
                                   int nnum, const int* __restrict__ depth_p,
                                   unsigned long long ws_bytes) {
    int depth = *depth_p;
    if (3 * depth > 30) return;
    long long n = 1ll << (3 * depth);
    if ((unsigned long long)n * 4ull > ws_bytes) return;
    int i = blockIdx.x * blockDim.x + threadIdx.x;
    if (i < nnum) {
        int k = keys[i];
        if (k >= 0 && (long long)k < n) inv[k] = i;   // keys unique -> no race
    }
}

// ---------------- main interpolation kernel ---------------------------------
__global__ __launch_bounds__(TPB) void octree_interp_kernel(
    const float* __restrict__ data, const float* __restrict__ pts,
    const int* __restrict__ keys, const int* __restrict__ depth_p,
    const int* __restrict__ inv, float* __restrict__ out,
    int npts, int nnum, unsigned long long ws_bytes)
{
    __shared__ int s_coarse[COARSE_MAX];

    const int depth = *depth_p;
    const int R = 1 << depth;
    const bool small_key = (3 * depth <= 30);
    const int lut_n = small_key ? (1 << (3 * depth)) : 0;
    const bool use_lut = small_key &&
                         ((unsigned long long)lut_n * 4ull <= ws_bytes);

    // Fallback path: stage a coarse search index (every strd-th key) into LDS
    // via CDNA5 async global->LDS loads, tracked with ASYNCcnt.
    int strd = 1, M = nnum;
    if (!use_lut) {
        while (M > COARSE_MAX) { strd <<= 1; M = (nnum + strd - 1) / strd; }
        for (int j = threadIdx.x; j < M; j += blockDim.x) {
            uint32_t loff  = lds_offset(&s_coarse[j]);
            uint64_t gaddr = (uint64_t)(uintptr_t)(keys + (size_t)j * strd);
            asm volatile("global_load_async_to_lds_b32 %0, %1, off"
                         :: "v"(loff), "v"(gaddr) : "memory");
        }
        asm volatile("s_wait_asynccnt 0" ::: "memory");
        __syncthreads();
    }

    const float scale = 0.5f * (float)R;           // 2^(depth-1)
    const int gstride = gridDim.x * blockDim.x;

    for (int i = blockIdx.x * blockDim.x + threadIdx.x; i < npts; i += gstride) {
        // pts is read-once (8 MB): non-temporal load keeps L2 for data+LUT.
        v4f p = __builtin_nontemporal_load(((const v4f*)pts) + i);
        float fx = (p.x + 1.0f) * scale - 0.5f;
        float fy = (p.y + 1.0f) * scale - 0.5f;
        float fz = (p.z + 1.0f) * scale - 0.5f;
        float flx = floorf(fx), fly = floorf(fy), flz = floorf(fz);
        int ix = (int)flx, iy = (int)fly, iz = (int)flz;
        float tx = fx - flx, ty = fy - fly, tz = fz - flz;
        float wxa[2] = {1.0f - tx, tx};
        float wya[2] = {1.0f - ty, ty};
        float wza[2] = {1.0f - tz, tz};
        int bb = (int)p.w;

        int   idx8[8];
        float w8[8];
        #pragma unroll
        for (int k = 0; k < 8; ++k) {
            int gx = (k >> 2) & 1, gy = (k >> 1) & 1, gz = k & 1;
            int x = ix + gx, y = iy + gy, z = iz + gz;
            bool inb = ((unsigned)x < (unsigned)R) & ((unsigned)y < (unsigned)R) &
                       ((unsigned)z < (unsigned)R);
            // int32 wrapping key math, matching jnp's int32 semantics exactly
            int key = (((bb * R + x) * R + y) * R + z);
            int idx = -1;
            if (inb) {
                if (use_lut) {
                    if ((unsigned)key < (unsigned)lut_n) idx = inv[key];  // 1 L2 load
                } else {
                    // coarse: first j with s_coarse[j] > key (LDS, ~log2(M) steps)
                    int lo = 0, hi = M;
                    while (lo < hi) {
                        int mid = (lo + hi) >> 1;
                        if (s_coarse[mid] <= key) lo = mid + 1; else hi = mid;
                    }
                    int a   = (lo == 0) ? 0 : (lo - 1) * strd;
                    int bnd = lo * strd; if (bnd > nnum) bnd = nnum;
                    // refine: log2(strd) dependent L2 loads
                    while (a < bnd) {
                        int mid = (a + bnd) >> 1;
                        if (keys[mid] < key) a = mid + 1; else bnd = mid;
                    }
                    if (a < nnum && keys[a] == key) idx = a;
                }
            }
            bool valid = idx > -1;
            w8[k]   = valid ? (wxa[gx] * wya[gy] * wza[gz]) : 0.0f;
            idx8[k] = valid ? idx : 0;                 // matches reference gather-at-0
        }

        v4f acc[8];
        #pragma unroll
        for (int q = 0; q < 8; ++q) acc[q] = (v4f){0.f, 0.f, 0.f, 0.f};
        float wsum = 0.0f;

        // 8 rows x 128B, each row = 8x global_load_b128, fully unrolled so all
        // gathers are in flight before the FMA chain consumes them.
        #pragma unroll
        for (int k = 0; k < 8; ++k) {
            const v4f* row = (const v4f*)(data + (size_t)idx8[k] * 32);
            float w = w8[k];
            wsum += w;
            #pragma unroll
            for (int q = 0; q < 8; ++q) {
                v4f r = row[q];
                acc[q] += w * r;
            }
        }

        float invn = 1.0f / (wsum + 1e-12f);
        // out is 64 MB write-once: stream with non-temporal stores so it does
        // not evict the L2-resident feature table.
        v4f* orow = (v4f*)(out + (size_t)i * 32);
        #pragma unroll
        for (int q = 0; q < 8; ++q)
            __builtin_nontemporal_store(acc[q] * invn, orow + q);
    }
}

// ---------------- host launcher ---------------------------------------------
extern "C" void kernel_launch(void* const* d_in, const int* in_sizes, int n_in,
                              void* d_out, int out_size, void* d_ws, size_t ws_size,
                              hipStream_t stream) {
    const float* data    = (const float*)d_in[0];   // [nnum, 32] f32
    const float* pts     = (const float*)d_in[1];   // [npts, 4]  f32
    const int*   keys    = (const int*)d_in[2];     // [nnum]     i32 sorted
    const int*   depth_p = (const int*)d_in[3];     // scalar
    float*       out     = (float*)d_out;           // [npts, 32] f32
    int*         inv     = (int*)d_ws;              // R^3 ints if it fits

    int nnum = in_sizes[2];
    int npts = in_sizes[1] / 4;
    unsigned long long wsb = (unsigned long long)ws_size;

    // Build inverse key->index LUT (kernels self-disable if d_ws is too small;
    // the same ws_size/depth test in the main kernel keeps the paths consistent).
    octree_fill_lut<<<2048, TPB, 0, stream>>>(inv, depth_p, wsb);
    octree_scatter_lut<<<(nnum + TPB - 1) / TPB, TPB, 0, stream>>>(inv, keys, nnum,
                                                                   depth_p, wsb);

    int grid = (npts + TPB - 1) / TPB;
    octree_interp_kernel<<<grid, TPB, 0, stream>>>(data, pts, keys, depth_p, inv,
                                                   out, npts, nnum, wsb);
}